// MultiRoundLSHAttention_32469952757890
// MI455X (gfx1250) — compile-verified
//
#include <hip/hip_runtime.h>
#include <hip/hip_bf16.h>

#define L_SEQ 4096
#define DMODEL 512
#define NHEAD 8
#define DK 64
#define NB 128        // buckets
#define NB2 64        // chunks per (bh, round)
#define CHUNK 64      // 2*bl
#define WIN 128       // 4*bl lookback window
#define RND 4
#define BH 16         // B*HEAD
#define BATCH 2
#define SENT 1000000000

typedef float v2f __attribute__((ext_vector_type(2)));
typedef float v8f __attribute__((ext_vector_type(8)));

// D = A(16x4 f32) x B(4x16 f32) + C  -> v_wmma_f32_16x16x4_f32
static __device__ __forceinline__ v8f wmma4(v2f a, v2f b, v8f c) {
  return __builtin_amdgcn_wmma_f32_16x16x4_f32(false, a, false, b, (short)0, c,
                                               false, false);
}

// ---------------------------------------------------------------------------
// C[m][n] = sum_k X[m][k] * W[n][k] + bias[n]    (M=8192, N=K=512)
// mode 0: store head-split  out[((b*8+h)*L + l)*64 + d]
// mode 1: store flat        out[m*512 + n]
// ---------------------------------------------------------------------------
__global__ void __launch_bounds__(256)
gemm512_wmma(const float* __restrict__ X, const float* __restrict__ W,
             const float* __restrict__ bias, float* __restrict__ out, int mode) {
  int wave = threadIdx.x >> 5, lane = threadIdx.x & 31;
  int m0 = blockIdx.x * 16;
  int n0 = (blockIdx.y * 8 + wave) * 16;
  int mrow = m0 + (lane & 15);
  int ncol = n0 + (lane & 15);
  int kh = (lane >= 16) ? 2 : 0;
  const float* xr = X + (size_t)mrow * DMODEL;
  const float* wr = W + (size_t)ncol * DMODEL;
  v8f acc = {};
  for (int k0 = 0; k0 < DMODEL; k0 += 4) {
    v2f a, b;
    a.x = xr[k0 + kh];  a.y = xr[k0 + kh + 1];
    b.x = wr[k0 + kh];  b.y = wr[k0 + kh + 1];
    acc = wmma4(a, b, acc);
  }
  int hi = (lane >= 16) ? 8 : 0;
  float bv = bias[ncol];
  for (int g = 0; g < 8; ++g) {
    int rm_ = m0 + g + hi;
    float val = acc[g] + bv;
    if (mode == 0) {
      int b_ = rm_ >> 12, l = rm_ & (L_SEQ - 1);
      int h = ncol >> 6, d = ncol & 63;
      out[(((size_t)(b_ * NHEAD + h) * L_SEQ) + l) * DK + d] = val;
    } else {
      out[(size_t)rm_ * DMODEL + ncol] = val;
    }
  }
}

// ---------------------------------------------------------------------------
__global__ void normalize_rows64(float* q, int nrows) {
  int i = blockIdx.x * blockDim.x + threadIdx.x;
  if (i >= nrows) return;
  float* row = q + (size_t)i * DK;
  float s = 0.f;
  for (int d = 0; d < DK; ++d) s += row[d] * row[d];
  float inv = rsqrtf(s);
  for (int d = 0; d < DK; ++d) row[d] *= inv;
}

__global__ void normalize_rand(const float* __restrict__ rm,
                               float* __restrict__ rmn) {
  int t = blockIdx.x * blockDim.x + threadIdx.x;  // (i, r) rows: 64*4
  if (t >= DK * RND) return;
  int i = t >> 2, r = t & 3;
  const float* src = rm + (size_t)i * (RND * (NB / 2)) + r * (NB / 2);
  float s = 0.f;
  for (int k = 0; k < NB / 2; ++k) s += src[k] * src[k];
  float inv = rsqrtf(s);
  float* dst = rmn + (size_t)i * (RND * (NB / 2)) + r * (NB / 2);
  for (int k = 0; k < NB / 2; ++k) dst[k] = src[k] * inv;
}

// ---------------------------------------------------------------------------
// hash = argmax over [proj, -proj]; first-max-wins tie handling (strict >)
__global__ void lsh_hash(const float* __restrict__ qn,
                         const float* __restrict__ rmn,
                         int* __restrict__ hashes) {
  int tid = blockIdx.x * blockDim.x + threadIdx.x;
  if (tid >= BH * L_SEQ * RND) return;
  int r = tid & 3;
  int l = (tid >> 2) & (L_SEQ - 1);
  int bh = tid >> 14;
  const float* q = qn + ((size_t)bh * L_SEQ + l) * DK;
  float best = -3.4e38f; int bi = 0;
  for (int k = 0; k < NB / 2; ++k) {
    float p = 0.f;
    for (int i = 0; i < DK; ++i) p += q[i] * rmn[i * (RND * (NB / 2)) + r * (NB / 2) + k];
    if (p > best) { best = p; bi = k; }
  }
  for (int k = 0; k < NB / 2; ++k) {
    float p = 0.f;
    for (int i = 0; i < DK; ++i) p += q[i] * rmn[i * (RND * (NB / 2)) + r * (NB / 2) + k];
    if (-p > best) { best = -p; bi = (NB / 2) + k; }
  }
  hashes[((size_t)bh * RND + r) * L_SEQ + l] = bi;
}

// ---------------------------------------------------------------------------
// stable counting sort per (bh, round): thread t == bucket t
__global__ void __launch_bounds__(128)
counting_sort(const int* __restrict__ hashes, int* __restrict__ hidx,
              int* __restrict__ shash, int* __restrict__ oidx) {
  int kb = blockIdx.x;  // bh*RND + r
  const int* h = hashes + (size_t)kb * L_SEQ;
  int t = threadIdx.x;
  int cnt = 0;
  for (int l = 0; l < L_SEQ; ++l) cnt += (h[l] == t);
  __shared__ int counts[NB];
  __shared__ int offs[NB];
  counts[t] = cnt;
  __syncthreads();
  if (t == 0) { int s = 0; for (int i = 0; i < NB; ++i) { offs[i] = s; s += counts[i]; } }
  __syncthreads();
  int pos = offs[t];
  int* hix = hidx  + (size_t)kb * L_SEQ;
  int* shx = shash + (size_t)kb * L_SEQ;
  int* oix = oidx  + (size_t)kb * L_SEQ;
  for (int l = 0; l < L_SEQ; ++l)
    if (h[l] == t) { hix[pos] = l; shx[pos] = t; oix[l] = pos; ++pos; }
}

// ---------------------------------------------------------------------------
// scores per (bh, chunk, round): S64x128 = Qs . Ks^T / 8, masks, scatter to
// original query order: S[bh][orig_l][jj][r]
__global__ void __launch_bounds__(128)
lsh_scores(const float* __restrict__ qn, const int* __restrict__ hidx,
           const int* __restrict__ shash, float* __restrict__ S) {
  __shared__ float Qs[CHUNK][DK];
  __shared__ float Ks[WIN][DK];
  __shared__ int qi[CHUNK], ki[WIN], shq[CHUNK], shk[WIN];
  int bid = blockIdx.x;
  int r = bid & 3, c = (bid >> 2) & 63, bh = bid >> 8;
  int sbase = (bh * RND + r) * L_SEQ;
  int t = threadIdx.x;
  if (t < CHUNK) { qi[t] = hidx[sbase + c * CHUNK + t]; shq[t] = shash[sbase + c * CHUNK + t]; }
  if (t < WIN) {
    int src = (t < CHUNK) ? ((c == 0) ? -1 : sbase + (c - 1) * CHUNK + t)
                          : sbase + c * CHUNK + (t - CHUNK);
    if (src < 0) { ki[t] = SENT; shk[t] = SENT; }
    else { ki[t] = hidx[src]; shk[t] = shash[src]; }
  }
  __syncthreads();
  size_t qbase = (size_t)bh * L_SEQ;
  for (int idx = t; idx < CHUNK * DK; idx += 128) {
    int i = idx >> 6, d = idx & 63;
    Qs[i][d] = qn[(qbase + qi[i]) * DK + d];
  }
  for (int idx = t; idx < WIN * DK; idx += 128) {
    int j = idx >> 6, d = idx & 63;
    Ks[j][d] = (ki[j] == SENT) ? 0.f : qn[(qbase + ki[j]) * DK + d];
  }
  __syncthreads();
  int wave = t >> 5, lane = t & 31;
  int m0 = wave * 16;
  int kh = (lane >= 16) ? 2 : 0;
  int hi = (lane >= 16) ? 8 : 0;
  int mrow = m0 + (lane & 15);
  for (int nt = 0; nt < WIN / 16; ++nt) {
    int ncol = nt * 16 + (lane & 15);
    v8f acc = {};
    for (int k0 = 0; k0 < DK; k0 += 4) {
      v2f a, b;
      a.x = Qs[mrow][k0 + kh];  a.y = Qs[mrow][k0 + kh + 1];
      b.x = Ks[ncol][k0 + kh];  b.y = Ks[ncol][k0 + kh + 1];
      acc = wmma4(a, b, acc);
    }
    for (int g = 0; g < 8; ++g) {
      int i = m0 + g + hi;
      int j = ncol;
      float sc = acc[g] * 0.125f;           // 1/sqrt(64)
      int kj = ki[j];
      if (kj == SENT) sc -= 3.0e9f;         // pad: fm + heq + causal
      else {
        if (shq[i] != shk[j]) sc -= 1.0e9f; // same-hash mask
        if (qi[i] <  kj)      sc -= 1.0e9f; // causal
        if (qi[i] == kj)      sc -= 1.0e5f; // self
      }
      S[((qbase + qi[i]) * WIN + j) * RND + r] = sc;
    }
  }
}

// ---------------------------------------------------------------------------
// per original query row: duplicate-key count correction (faithful loop),
// softmax over 512; one wave32 per row, in place
__global__ void __launch_bounds__(256)
lsh_softmax(float* __restrict__ S, const int* __restrict__ hidx,
            const int* __restrict__ oidx) {
  int wave = threadIdx.x >> 5, lane = threadIdx.x & 31;
  int row = blockIdx.x * 8 + wave;           // bh*L_SEQ rows
  int bh = row >> 12, l = row & (L_SEQ - 1);
  size_t sb = (size_t)row * (WIN * RND);
  float vals[16];
  for (int jq = 0; jq < 4; ++jq) {
    int jj = lane * 4 + jq;
    int rk[RND];
    for (int r = 0; r < RND; ++r) {
      int p = oidx[(bh * RND + r) * L_SEQ + l];
      int c = p >> 6;
      if (jj < CHUNK) rk[r] = (c == 0) ? SENT
                              : hidx[(bh * RND + r) * L_SEQ + (c - 1) * CHUNK + jj];
      else            rk[r] = hidx[(bh * RND + r) * L_SEQ + c * CHUNK + (jj - CHUNK)];
    }
    int cnt[RND] = {1, 1, 1, 1};
    for (int i2 = 0; i2 < RND - 1; ++i2)
      for (int j2 = 0; j2 < RND - 1 - i2; ++j2) {
        int comp = (rk[i2] == rk[i2 + 1 + j2]) ? 1 : 0;
        cnt[i2] += comp;
        cnt[j2] += comp;                      // faithful to reference's indexing
      }
    for (int r = 0; r < RND; ++r)
      vals[jq * 4 + r] = S[sb + (size_t)jj * RND + r] - __logf((float)cnt[r]);
  }
  float mx = vals[0];
  for (int e = 1; e < 16; ++e) mx = fmaxf(mx, vals[e]);
  for (int off = 16; off > 0; off >>= 1) mx = fmaxf(mx, __shfl_xor(mx, off, 32));
  float sum = 0.f;
  for (int e = 0; e < 16; ++e) { vals[e] = __expf(vals[e] - mx); sum += vals[e]; }
  for (int off = 16; off > 0; off >>= 1) sum += __shfl_xor(sum, off, 32);
  float inv = 1.f / sum;
  for (int jq = 0; jq < 4; ++jq) {
    int jj = lane * 4 + jq;
    for (int r = 0; r < RND; ++r)
      S[sb + (size_t)jj * RND + r] = vals[jq * 4 + r] * inv;
  }
}

// ---------------------------------------------------------------------------
// O64x64 = P64x128 . V128x64 per (bh, chunk, round); scatter per-round output
__global__ void __launch_bounds__(128)
lsh_av(const float* __restrict__ S, const float* __restrict__ vp,
       const int* __restrict__ hidx, float* __restrict__ O4) {
  __shared__ float P[CHUNK][WIN];
  __shared__ int qi[CHUNK], ki[WIN];
  int bid = blockIdx.x;
  int r = bid & 3, c = (bid >> 2) & 63, bh = bid >> 8;
  int sbase = (bh * RND + r) * L_SEQ;
  int t = threadIdx.x;
  if (t < CHUNK) qi[t] = hidx[sbase + c * CHUNK + t];
  if (t < WIN) {
    if (t < CHUNK) ki[t] = (c == 0) ? SENT : hidx[sbase + (c - 1) * CHUNK + t];
    else           ki[t] = hidx[sbase + c * CHUNK + (t - CHUNK)];
  }
  __syncthreads();
  size_t qbase = (size_t)bh * L_SEQ;
  for (int idx = t; idx < CHUNK * WIN; idx += 128) {
    int i = idx >> 7, jj = idx & 127;
    P[i][jj] = S[((qbase + qi[i]) * WIN + jj) * RND + r];
  }
  __syncthreads();
  int wave = t >> 5, lane = t & 31;
  int m0 = wave * 16;
  int kh = (lane >= 16) ? 2 : 0;
  int hi = (lane >= 16) ? 8 : 0;
  int mrow = m0 + (lane & 15);
  for (int nt = 0; nt < DK / 16; ++nt) {
    int ncol = nt * 16 + (lane & 15);
    v8f acc = {};
    for (int k0 = 0; k0 < WIN; k0 += 4) {
      int kA = k0 + kh;
      v2f a, b;
      a.x = P[mrow][kA];  a.y = P[mrow][kA + 1];
      int k1 = ki[kA], k2 = ki[kA + 1];
      b.x = (k1 == SENT) ? 0.f : vp[(qbase + k1) * DK + ncol];
      b.y = (k2 == SENT) ? 0.f : vp[(qbase + k2) * DK + ncol];
      acc = wmma4(a, b, acc);
    }
    for (int g = 0; g < 8; ++g) {
      int i = m0 + g + hi;
      O4[(((size_t)(bh * RND + r)) * L_SEQ + qi[i]) * DK + ncol] = acc[g];
    }
  }
}

// ---------------------------------------------------------------------------
__global__ void sum_rounds_concat(const float* __restrict__ O4,
                                  float* __restrict__ xcat) {
  int idx = blockIdx.x * blockDim.x + threadIdx.x;
  if (idx >= BATCH * L_SEQ * DMODEL) return;
  int n = idx & (DMODEL - 1);
  int l = (idx >> 9) & (L_SEQ - 1);
  int b = idx >> 21;
  int h = n >> 6, d = n & 63;
  int bh = b * NHEAD + h;
  float s = 0.f;
  for (int r = 0; r < RND; ++r)
    s += O4[(((size_t)(bh * RND + r)) * L_SEQ + l) * DK + d];
  xcat[idx] = s;
}

// ---------------------------------------------------------------------------
extern "C" void kernel_launch(void* const* d_in, const int* in_sizes, int n_in,
                              void* d_out, int out_size, void* d_ws,
                              size_t ws_size, hipStream_t stream) {
  (void)in_sizes; (void)n_in; (void)out_size; (void)ws_size;
  const float* query = (const float*)d_in[0];
  const float* value = (const float*)d_in[1];
  // d_in[2] = mask (all true in this problem; padding handled via lookback pad)
  const float* Wq = (const float*)d_in[3];
  const float* bq = (const float*)d_in[4];
  const float* Wv = (const float*)d_in[5];
  const float* bv = (const float*)d_in[6];
  const float* Wo = (const float*)d_in[7];
  const float* bo = (const float*)d_in[8];
  const float* rm = (const float*)d_in[9];

  char* ws = (char*)d_ws;
  size_t off = 0;
  auto alloc = [&](size_t bytes) -> void* {
    void* p = ws + off;
    off += (bytes + 255) & ~(size_t)255;
    return p;
  };
  float* qn    = (float*)alloc((size_t)BH * L_SEQ * DK * 4);           // 16 MB
  float* vp    = (float*)alloc((size_t)BH * L_SEQ * DK * 4);           // 16 MB
  float* rmn   = (float*)alloc((size_t)DK * RND * (NB / 2) * 4);       // 64 KB
  int*   hashes= (int*)  alloc((size_t)BH * RND * L_SEQ * 4);          // 1 MB
  int*   hidx  = (int*)  alloc((size_t)BH * RND * L_SEQ * 4);          // 1 MB
  int*   shash = (int*)  alloc((size_t)BH * RND * L_SEQ * 4);          // 1 MB
  int*   oidx  = (int*)  alloc((size_t)BH * RND * L_SEQ * 4);          // 1 MB
  float* S     = (float*)alloc((size_t)BH * L_SEQ * WIN * RND * 4);    // 128 MB
  float* O4    = (float*)alloc((size_t)BH * RND * L_SEQ * DK * 4);     // 64 MB
  float* xcat  = (float*)alloc((size_t)BATCH * L_SEQ * DMODEL * 4);    // 16 MB

  const int M = BATCH * L_SEQ;
  dim3 gblk(256);
  dim3 ggrid(M / 16, DMODEL / 128);

  gemm512_wmma<<<ggrid, gblk, 0, stream>>>(query, Wq, bq, qn, 0);
  gemm512_wmma<<<ggrid, gblk, 0, stream>>>(value, Wv, bv, vp, 0);
  normalize_rows64<<<(BH * L_SEQ) / 256, 256, 0, stream>>>(qn, BH * L_SEQ);
  normalize_rand<<<1, 256, 0, stream>>>(rm, rmn);
  lsh_hash<<<(BH * L_SEQ * RND) / 256, 256, 0, stream>>>(qn, rmn, hashes);
  counting_sort<<<BH * RND, 128, 0, stream>>>(hashes, hidx, shash, oidx);
  lsh_scores<<<BH * NB2 * RND, 128, 0, stream>>>(qn, hidx, shash, S);
  lsh_softmax<<<(BH * L_SEQ) / 8, 256, 0, stream>>>(S, hidx, oidx);
  lsh_av<<<BH * NB2 * RND, 128, 0, stream>>>(S, vp, hidx, O4);
  sum_rounds_concat<<<(BATCH * L_SEQ * DMODEL) / 256, 256, 0, stream>>>(O4, xcat);
  gemm512_wmma<<<ggrid, gblk, 0, stream>>>(xcat, Wo, bo, (float*)d_out, 1);
}